// RegularizedMeshLoss_38680475467969
// MI455X (gfx1250) — compile-verified
//
#include <hip/hip_runtime.h>
#include <hip/hip_bf16.h>
#include <cstdint>
#include <cstddef>

// ---------------- problem constants (from reference) ----------------
#define BATCH   8
#define NSAMP   2048          // N_SAMPLES
#define BN      (BATCH * NSAMP)
#define W_NORM  0.1f
#define W_LAP   0.1f

typedef __attribute__((ext_vector_type(2)))  float    v2f;
typedef __attribute__((ext_vector_type(8)))  float    v8f;

// ---------------- small helpers ----------------
__device__ __forceinline__ unsigned hash32(unsigned x) {
  x ^= x >> 16; x *= 0x7feb352dU;
  x ^= x >> 15; x *= 0x846ca68bU;
  x ^= x >> 16; return x;
}
__device__ __forceinline__ float u01(unsigned h) {
  return (float)(h >> 8) * (1.0f / 16777216.0f);
}

__device__ __forceinline__ float blockReduceAdd(float v, float* sm) {
  sm[threadIdx.x] = v;
  __syncthreads();
  for (int o = blockDim.x >> 1; o > 0; o >>= 1) {
    if ((int)threadIdx.x < o) sm[threadIdx.x] += sm[threadIdx.x + o];
    __syncthreads();
  }
  return sm[0];
}

// ---------------- workspace init (harness poisons ws with 0xAA) ----------------
__global__ void init_ws(unsigned* __restrict__ minP, unsigned* __restrict__ minT,
                        float* __restrict__ nsum, float* __restrict__ deg,
                        float* __restrict__ acc, size_t nsumN, int V) {
  size_t i = (size_t)blockIdx.x * blockDim.x + threadIdx.x;
  size_t stride = (size_t)gridDim.x * blockDim.x;
  for (size_t k = i; k < (size_t)BN; k += stride) { minP[k] = 0x7F800000u; minT[k] = 0x7F800000u; }
  for (size_t k = i; k < nsumN; k += stride) nsum[k] = 0.0f;
  for (size_t k = i; k < (size_t)V; k += stride) deg[k] = 0.0f;
  if (i < 8) acc[i] = 0.0f;
}

// ---------------- surface sampling (deterministic hash RNG stand-in for threefry) ----------------
// Writes (x,y,z,|p|^2) per sample so the chamfer kernel only needs norms + WMMA dot products.
__global__ void sample_points(const float* __restrict__ vp, const float* __restrict__ vt,
                              const int* __restrict__ faces, int F, int V,
                              float* __restrict__ sampP, float* __restrict__ sampT) {
  int idx = blockIdx.x * blockDim.x + threadIdx.x;
  if (idx >= 2 * BN) return;
  int which = idx / BN;            // 0 = pred, 1 = targ
  int r     = idx % BN;            // b*NSAMP + s
  int b     = r / NSAMP;
  const float* verts = (which ? vt : vp) + (size_t)b * V * 3;

  unsigned h0 = hash32((unsigned)idx * 0x9E3779B9u + 0x01234567u);
  unsigned h1 = hash32(h0 ^ 0xABCDEF01u);
  unsigned h2 = hash32(h1 + 0x68bc21ebu);
  int f = (int)(h0 % (unsigned)F);
  float su = sqrtf(u01(h1));
  float vv = u01(h2);
  float w0 = 1.0f - su, w1 = su * (1.0f - vv), w2 = su * vv;

  int i0 = faces[f * 3 + 0], i1 = faces[f * 3 + 1], i2 = faces[f * 3 + 2];
  float x = w0 * verts[i0 * 3 + 0] + w1 * verts[i1 * 3 + 0] + w2 * verts[i2 * 3 + 0];
  float y = w0 * verts[i0 * 3 + 1] + w1 * verts[i1 * 3 + 1] + w2 * verts[i2 * 3 + 1];
  float z = w0 * verts[i0 * 3 + 2] + w1 * verts[i1 * 3 + 2] + w2 * verts[i2 * 3 + 2];

  float* out = (which ? sampT : sampP) + (size_t)r * 4;
  out[0] = x; out[1] = y; out[2] = z; out[3] = x * x + y * y + z * z;
}

// ---------------- chamfer: LDS-resident target sweep, WMMA f32 16x16x4 ----------------
// One block = 8 waves = 128 pred rows of one batch. The batch's whole target set
// (2048 x float4 = 32 KB) is staged in LDS once; each wave keeps its A-fragment and
// row-min accumulators in registers across all 128 column tiles (128 WMMAs/wave).
// Row mins: register-resident, shuffle-reduced, plain stores (row owned by one wave).
// Col mins: ds-atomicMin into LDS colmin[], one global atomicMin pass per block.
__global__ void __launch_bounds__(256)
chamfer_sweep(const float* __restrict__ sampP, const float* __restrict__ sampT,
              unsigned* __restrict__ minP, unsigned* __restrict__ minT) {
  __shared__ float4   ldsT[NSAMP];      // 32 KB
  __shared__ unsigned colmin[NSAMP];    //  8 KB
  const int tilesPerSide = NSAMP / 16;  // 128

  const int b        = blockIdx.x >> 4;     // 16 blocks per batch
  const int rowChunk = blockIdx.x & 15;
  const int tid  = threadIdx.x;
  const int wave = tid >> 5;
  const int lane = tid & 31;
  const int half = lane >> 4;               // 0 -> K=0,1 (x,y) ; 1 -> K=2,3 (z,0)
  const int lrow = lane & 15;

  const float4* Tb = (const float4*)(sampT + (size_t)b * NSAMP * 4);
  const float4* Pb = (const float4*)(sampP + (size_t)b * NSAMP * 4);

  for (int i = tid; i < NSAMP; i += 256) { ldsT[i] = Tb[i]; colmin[i] = 0x7F800000u; }
  __syncthreads();

  const int ti = rowChunk * 8 + wave;       // this wave's pred row tile
  __builtin_prefetch(Pb + ti * 16, 0, 0);   // global_prefetch_b8

  // A-fragment per ISA 16x4 f32 layout
  float4 ap = Pb[ti * 16 + lrow];
  v2f afrag;
  afrag.x = half ? ap.z : ap.x;
  afrag.y = half ? 0.0f : ap.y;

  float pn[8];
#pragma unroll
  for (int r = 0; r < 8; ++r) pn[r] = Pb[ti * 16 + r + 8 * half].w;

  float rmin[8];
#pragma unroll
  for (int r = 0; r < 8; ++r) rmin[r] = 3.4e38f;

#pragma unroll 4
  for (int tj = 0; tj < tilesPerSide; ++tj) {
    float4 tp = ldsT[tj * 16 + lrow];
    v2f bfrag;
    bfrag.x = half ? tp.z : tp.x;
    bfrag.y = half ? 0.0f : tp.y;
    float tn = tp.w;

    v8f c = {};
    c = __builtin_amdgcn_wmma_f32_16x16x4_f32(false, afrag, false, bfrag,
                                              (short)0, c, false, false);

    float cmin = 3.4e38f;
#pragma unroll
    for (int r = 0; r < 8; ++r) {
      float d = fmaxf(pn[r] + tn - 2.0f * c[r], 0.0f);
      rmin[r] = fminf(rmin[r], d);
      cmin    = fminf(cmin, d);
    }
    cmin = fminf(cmin, __shfl_xor(cmin, 16, 32));     // combine the two 8-row halves
    if (half == 0)
      atomicMin(&colmin[tj * 16 + lrow], __float_as_uint(cmin));
  }

  // row mins: reduce across the 16 lanes of each half; plain store (no atomics needed)
#pragma unroll
  for (int r = 0; r < 8; ++r) {
    float v = rmin[r];
    v = fminf(v, __shfl_xor(v, 1, 32));
    v = fminf(v, __shfl_xor(v, 2, 32));
    v = fminf(v, __shfl_xor(v, 4, 32));
    v = fminf(v, __shfl_xor(v, 8, 32));
    if (lrow == 0)
      minP[(size_t)b * NSAMP + ti * 16 + r + 8 * half] = __float_as_uint(v);
  }

  __syncthreads();
  for (int i = tid; i < NSAMP; i += 256)
    atomicMin(&minT[(size_t)b * NSAMP + i], colmin[i]);
}

// ---------------- reductions for the remaining loss terms ----------------
__global__ void reduce_min_sums(const unsigned* __restrict__ minP,
                                const unsigned* __restrict__ minT,
                                float* __restrict__ acc) {
  __shared__ float sm[256];
  const unsigned* src = blockIdx.y ? minT : minP;
  float s = 0.0f;
  for (int i = blockIdx.x * blockDim.x + threadIdx.x; i < BN; i += gridDim.x * blockDim.x)
    s += __uint_as_float(src[i]);
  float t = blockReduceAdd(s, sm);
  if (threadIdx.x == 0) atomicAdd(&acc[blockIdx.y], t);   // acc[0]=pred-side, acc[1]=targ-side
}

__global__ void edge_loss_k(const float* __restrict__ vp, const int* __restrict__ edges,
                            int E, int V, float* __restrict__ acc) {
  __shared__ float sm[256];
  int total = BATCH * E;
  float s = 0.0f;
  for (int i = blockIdx.x * blockDim.x + threadIdx.x; i < total; i += gridDim.x * blockDim.x) {
    int b = i / E, e = i % E;
    const float* base = vp + (size_t)b * V * 3;
    int a0 = edges[e * 2 + 0], a1 = edges[e * 2 + 1];
    float dx = base[a0 * 3 + 0] - base[a1 * 3 + 0];
    float dy = base[a0 * 3 + 1] - base[a1 * 3 + 1];
    float dz = base[a0 * 3 + 2] - base[a1 * 3 + 2];
    s += dx * dx + dy * dy + dz * dz;
  }
  float t = blockReduceAdd(s, sm);
  if (threadIdx.x == 0) atomicAdd(&acc[2], t);
}

__global__ void normal_loss_k(const float* __restrict__ vp, const int* __restrict__ ncp,
                              int P, int V, float* __restrict__ acc) {
  __shared__ float sm[256];
  int total = BATCH * P;
  float s = 0.0f;
  for (int i = blockIdx.x * blockDim.x + threadIdx.x; i < total; i += gridDim.x * blockDim.x) {
    int b = i / P, p = i % P;
    const float* base = vp + (size_t)b * V * 3;
    int i0 = ncp[p * 4 + 0], i1 = ncp[p * 4 + 1], ia = ncp[p * 4 + 2], ib = ncp[p * 4 + 3];
    float v0x = base[i0 * 3 + 0], v0y = base[i0 * 3 + 1], v0z = base[i0 * 3 + 2];
    float ex  = base[i1 * 3 + 0] - v0x, ey = base[i1 * 3 + 1] - v0y, ez = base[i1 * 3 + 2] - v0z;
    float ax  = base[ia * 3 + 0] - v0x, ay = base[ia * 3 + 1] - v0y, az = base[ia * 3 + 2] - v0z;
    float bx  = base[ib * 3 + 0] - v0x, by = base[ib * 3 + 1] - v0y, bz = base[ib * 3 + 2] - v0z;
    // n0 = e x a ; n1 = -(e x b)
    float n0x = ey * az - ez * ay, n0y = ez * ax - ex * az, n0z = ex * ay - ey * ax;
    float n1x = -(ey * bz - ez * by), n1y = -(ez * bx - ex * bz), n1z = -(ex * by - ey * bx);
    float eps = 1e-8f;
    float l0 = fmaxf(sqrtf(n0x * n0x + n0y * n0y + n0z * n0z), eps);
    float l1 = fmaxf(sqrtf(n1x * n1x + n1y * n1y + n1z * n1z), eps);
    float cosv = (n0x * n1x + n0y * n1y + n0z * n1z) / (l0 * l1);
    s += 1.0f - cosv;
  }
  float t = blockReduceAdd(s, sm);
  if (threadIdx.x == 0) atomicAdd(&acc[3], t);
}

__global__ void lap_scatter(const float* __restrict__ vp, const int* __restrict__ edges,
                            int E, int V, float* __restrict__ nsum, float* __restrict__ deg) {
  int i = blockIdx.x * blockDim.x + threadIdx.x;
  if (i < E) {                       // degree is batch-independent
    atomicAdd(&deg[edges[i * 2 + 0]], 1.0f);
    atomicAdd(&deg[edges[i * 2 + 1]], 1.0f);
  }
  int total = BATCH * E;
  if (i >= total) return;
  int b = i / E, e = i % E;
  int a0 = edges[e * 2 + 0], a1 = edges[e * 2 + 1];
  const float* base = vp + (size_t)b * V * 3;
  float* nb = nsum + (size_t)b * V * 3;
#pragma unroll
  for (int c = 0; c < 3; ++c) {
    atomicAdd(&nb[a0 * 3 + c], base[a1 * 3 + c]);
    atomicAdd(&nb[a1 * 3 + c], base[a0 * 3 + c]);
  }
}

__global__ void lap_norm_k(const float* __restrict__ vp, const float* __restrict__ nsum,
                           const float* __restrict__ deg, int V, float* __restrict__ acc) {
  __shared__ float sm[256];
  int total = BATCH * V;
  float s = 0.0f;
  for (int i = blockIdx.x * blockDim.x + threadIdx.x; i < total; i += gridDim.x * blockDim.x) {
    int b = i / V, v = i % V;
    const float* base = vp + (size_t)b * V * 3 + (size_t)v * 3;
    const float* nb   = nsum + (size_t)b * V * 3 + (size_t)v * 3;
    float dinv = 1.0f / fmaxf(deg[v], 1.0f);
    float lx = nb[0] * dinv - base[0];
    float ly = nb[1] * dinv - base[1];
    float lz = nb[2] * dinv - base[2];
    s += sqrtf(lx * lx + ly * ly + lz * lz);
  }
  float t = blockReduceAdd(s, sm);
  if (threadIdx.x == 0) atomicAdd(&acc[4], t);
}

__global__ void finalize_k(const float* __restrict__ acc, float* __restrict__ out,
                           int E, int P, int V) {
  if (threadIdx.x == 0 && blockIdx.x == 0) {
    float bn      = (float)BN;
    float chamfer = acc[0] / bn + acc[1] / bn;
    float edge    = acc[2] / ((float)BATCH * (float)E);
    float normal  = acc[3] / ((float)BATCH * (float)P);
    float lap     = acc[4] / ((float)BATCH * (float)V);
    float loss    = chamfer + edge + W_NORM * normal + W_LAP * lap;
    out[0] = loss; out[1] = chamfer; out[2] = edge; out[3] = normal; out[4] = lap;
  }
}

// ---------------- host-side launcher ----------------
extern "C" void kernel_launch(void* const* d_in, const int* in_sizes, int n_in,
                              void* d_out, int out_size, void* d_ws, size_t ws_size,
                              hipStream_t stream) {
  (void)n_in; (void)out_size; (void)ws_size;
  const float* vp    = (const float*)d_in[0];   // verts_pred [B,V,3]
  const float* vt    = (const float*)d_in[1];   // verts_targ [B,V,3]
  const int*   faces = (const int*)d_in[2];     // [F,3]
  const int*   edges = (const int*)d_in[3];     // [E,2]
  const int*   ncp   = (const int*)d_in[4];     // [P,4]
  const int F = in_sizes[2] / 3;
  const int E = in_sizes[3] / 2;
  const int P = in_sizes[4] / 4;
  const int V = in_sizes[0] / (3 * BATCH);

  // workspace carve-up (~6.9 MB)
  char* ws = (char*)d_ws;
  float*    sampP = (float*)ws;    ws += (size_t)BN * 4 * sizeof(float);
  float*    sampT = (float*)ws;    ws += (size_t)BN * 4 * sizeof(float);
  unsigned* minP  = (unsigned*)ws; ws += (size_t)BN * sizeof(unsigned);
  unsigned* minT  = (unsigned*)ws; ws += (size_t)BN * sizeof(unsigned);
  float*    deg   = (float*)ws;    ws += (size_t)V * sizeof(float);
  float*    nsum  = (float*)ws;    ws += (size_t)BATCH * V * 3 * sizeof(float);
  float*    acc   = (float*)ws;    ws += 8 * sizeof(float);

  const size_t nsumN = (size_t)BATCH * V * 3;

  init_ws<<<4096, 256, 0, stream>>>(minP, minT, nsum, deg, acc, nsumN, V);

  sample_points<<<(2 * BN + 255) / 256, 256, 0, stream>>>(vp, vt, faces, F, V, sampP, sampT);

  // 16 row-chunks per batch (8 waves x 16 rows each), 8 batches -> 128 blocks
  chamfer_sweep<<<BATCH * 16, 256, 0, stream>>>(sampP, sampT, minP, minT);

  { dim3 g(64, 2); reduce_min_sums<<<g, 256, 0, stream>>>(minP, minT, acc); }

  edge_loss_k<<<1024, 256, 0, stream>>>(vp, edges, E, V, acc);
  normal_loss_k<<<1024, 256, 0, stream>>>(vp, ncp, P, V, acc);
  lap_scatter<<<(BATCH * E + 255) / 256, 256, 0, stream>>>(vp, edges, E, V, nsum, deg);
  lap_norm_k<<<1024, 256, 0, stream>>>(vp, nsum, deg, V, acc);

  finalize_k<<<1, 32, 0, stream>>>(acc, (float*)d_out, E, P, V);
}